// NCNNetTestCuda_20358144983218
// MI455X (gfx1250) — compile-verified
//
#include <hip/hip_runtime.h>
#include <hip/hip_bf16.h>

// Problem constants (from reference)
#define BB   8
#define TT   4096
#define DD   1024
#define HH   16
#define DH   64
#define NC   8
#define BT   (BB * TT)              // 32768 rows
#define NSTEP (TT / NC)             // 512 chain steps per slot

typedef __attribute__((ext_vector_type(2))) float v2f;
typedef __attribute__((ext_vector_type(8))) float v8f;

#define AS_STRIDE 68     // padded row stride for A tile in LDS (conflict-free b64 reads)
#define WP_STRIDE 160    // floats per K-pair row of W: 64 cols * 2 + 32 pad
                         // 160 % 64 == 32 -> lane halves hit disjoint bank halves

// ---------------------------------------------------------------------------
// GEMM: mix[r, h*64+e] = sum_k X[r, h*64+k] * W[h, k, e]
// grid = (BT/128, HH), block = 256 threads (8 waves), wave -> 16 rows x 64 cols
//
// W is staged pair-swizzled: sWp[p*WP_STRIDE + c*2 + j] = W[2p+j][c], so each
// B operand (v2f = K rows {2p, 2p+1} of one column) is a single ds_load_b64.
// ---------------------------------------------------------------------------
__global__ __launch_bounds__(256, 1)
void ncn_gemm(const float* __restrict__ X, const float* __restrict__ Wm,
              float* __restrict__ mix)
{
    __shared__ float sWp[32 * WP_STRIDE];       // W[h] 64x64, pair-swizzled
    __shared__ float sA[8 * 16 * AS_STRIDE];    // 128 rows x 64 cols of X, padded

    const int h    = blockIdx.y;
    const int row0 = blockIdx.x * 128;
    const int tid  = threadIdx.x;

    // Stage W[h] (row-major [k][e]) into LDS, pair-swizzled.
    const float* __restrict__ Wh = Wm + (size_t)h * 64 * 64;
    #pragma unroll
    for (int i = tid; i < 64 * 64; i += 256) {
        int k = i >> 6, c = i & 63;
        sWp[(k >> 1) * WP_STRIDE + c * 2 + (k & 1)] = Wh[i];
    }

    // Stage 128 rows x 64 cols of X into LDS via float4, fully coalesced.
    #pragma unroll
    for (int p = 0; p < 8; ++p) {
        int idx = p * 256 + tid;
        int r   = idx >> 4;              // 0..127
        int c4  = (idx & 15) << 2;       // 0,4,..,60
        const float4 v = *(const float4*)(X + (size_t)(row0 + r) * DD + h * 64 + c4);
        float* dst = &sA[r * AS_STRIDE + c4];
        dst[0] = v.x; dst[1] = v.y; dst[2] = v.z; dst[3] = v.w;
    }
    __syncthreads();

    const int wave = tid >> 5;
    const int lane = tid & 31;
    const int n    = lane & 15;          // A: M index, B/C/D: N index
    const int kh   = lane >> 4;          // lane-half selector

    const float* __restrict__ Arow = &sA[(wave * 16 + n) * AS_STRIDE];

    v8f acc0 = {}, acc1 = {}, acc2 = {}, acc3 = {};

    #pragma unroll
    for (int kk = 0; kk < 16; ++kk) {
        // A 16x4 f32: lanes 0-15 hold K={4kk,4kk+1}, lanes 16-31 K={4kk+2,4kk+3}
        const v2f a = *(const v2f*)&Arow[kk * 4 + kh * 2];

        // B 4x16 f32 per N-tile: pair row p selects K rows {2p, 2p+1};
        // one b64 per tile, bank-conflict-free.
        const float* __restrict__ Wp = &sWp[(kk * 2 + kh) * WP_STRIDE + n * 2];
        const v2f b0 = *(const v2f*)(Wp +  0 * 32);
        const v2f b1 = *(const v2f*)(Wp +  1 * 32);
        const v2f b2 = *(const v2f*)(Wp +  2 * 32);
        const v2f b3 = *(const v2f*)(Wp +  3 * 32);

        acc0 = __builtin_amdgcn_wmma_f32_16x16x4_f32(false, a, false, b0, (short)0, acc0, false, false);
        acc1 = __builtin_amdgcn_wmma_f32_16x16x4_f32(false, a, false, b1, (short)0, acc1, false, false);
        acc2 = __builtin_amdgcn_wmma_f32_16x16x4_f32(false, a, false, b2, (short)0, acc2, false, false);
        acc3 = __builtin_amdgcn_wmma_f32_16x16x4_f32(false, a, false, b3, (short)0, acc3, false, false);
    }

    // C/D 16x16 f32: lane gives N=n, VGPR v + 8*kh gives M. 16 consecutive lanes
    // write 16 consecutive floats -> coalesced 64B segments.
    float* __restrict__ out = mix + (size_t)(row0 + wave * 16) * DD + h * 64;
    #pragma unroll
    for (int v = 0; v < 8; ++v) {
        const size_t ro = (size_t)(v + 8 * kh) * DD;
        out[ro +  0 + n] = acc0[v];
        out[ro + 16 + n] = acc1[v];
        out[ro + 32 + n] = acc2[v];
        out[ro + 48 + n] = acc3[v];
    }
}

// ---------------------------------------------------------------------------
// Scan: per (b, slot, d) chain of length 512:
//   c = tanh(0.5*c + 0.5*mix[b, s + 8j, d]);  y[b, s+8j, d] = c
// 65536 threads; consecutive d per lane -> fully coalesced stride-8192 streams.
// ---------------------------------------------------------------------------
__global__ __launch_bounds__(256, 4)
void ncn_scan(const float* __restrict__ mix, float* __restrict__ y,
              float* __restrict__ cache)
{
    const int idx = blockIdx.x * 256 + threadIdx.x;   // 0 .. 65535
    const int d = idx & (DD - 1);
    const int s = (idx >> 10) & (NC - 1);
    const int b = idx >> 13;

    const size_t base = (size_t)b * TT * DD + (size_t)s * DD + d;
    const size_t cidx = (size_t)b * NC * DD + (size_t)s * DD + d;
    const size_t stride = (size_t)NC * DD;            // 8192 floats

    float c = cache[cidx];
    const float* __restrict__ mp = mix + base;
    float*       __restrict__ yp = y   + base;

    #pragma unroll 8
    for (int j = 0; j < NSTEP; ++j) {
        const float mv = mp[(size_t)j * stride];      // independent of chain -> issues ahead
        c = tanhf(0.5f * (c + mv));                   // ALPHA = 0.5
        yp[(size_t)j * stride] = c;
    }
    cache[cidx] = c;
}

// ---------------------------------------------------------------------------
// Launch: 4 modules, each = GEMM (x -> mix in ws) then scan (mix -> y, cache).
// yi region of d_out doubles as the inter-module x buffer; ya region of d_out
// holds the live cache (seeded from xa).
// ---------------------------------------------------------------------------
extern "C" void kernel_launch(void* const* d_in, const int* in_sizes, int n_in,
                              void* d_out, int out_size, void* d_ws, size_t ws_size,
                              hipStream_t stream) {
    const float* x  = (const float*)d_in[0];   // [B,T,D]
    const float* xa = (const float*)d_in[1];   // [B,NC,D]
    const float* w  = (const float*)d_in[2];   // [4,H,DH,DH]

    float* yi  = (float*)d_out;                      // [B,T,D]
    float* ya  = yi + (size_t)BB * TT * DD;          // [B,NC,D]
    float* mix = (float*)d_ws;                       // [B,T,D] scratch (134 MB, L2-resident)

    // Seed live cache with xa (device-to-device, graph-capture safe).
    hipMemcpyAsync(ya, xa, (size_t)BB * NC * DD * sizeof(float),
                   hipMemcpyDeviceToDevice, stream);

    dim3 ggrid(BT / 128, HH);
    const float* xcur = x;
    for (int m = 0; m < 4; ++m) {
        ncn_gemm<<<ggrid, 256, 0, stream>>>(xcur, w + (size_t)m * HH * DH * DH, mix);
        ncn_scan<<<(BB * NC * DD) / 256, 256, 0, stream>>>(mix, yi, ya);
        xcur = yi;   // next module consumes this module's output
    }
}